// DeepMUSIC_44555990729127
// MI455X (gfx1250) — compile-verified
//
#include <hip/hip_runtime.h>
#include <hip/hip_bf16.h>
#include <math.h>

typedef __attribute__((ext_vector_type(2))) float v2f;
typedef __attribute__((ext_vector_type(8))) float v8f;

#define N_MICS   16
#define T_LEN    8192
#define GRU_H    128
#define G3       (3 * GRU_H)      // 384
#define N_THETAS 65536
#define MLP_H    256

// ---------------------------------------------------------------------------
// 1) Per-mic normalize (mean, ddof=1 std) and write transposed f32 s.T
//    sT[t*16 + m] = (s[m,t]-mu_m)/sd_m
// ---------------------------------------------------------------------------
__global__ void norm_kernel(const float* __restrict__ samples,
                            float* __restrict__ sT) {
    __shared__ float red[256];
    const int r   = blockIdx.x;          // mic row 0..15
    const int tid = threadIdx.x;
    const float* row = samples + r * T_LEN;

    float acc = 0.f;
    for (int t = tid; t < T_LEN; t += 256) acc += row[t];
    red[tid] = acc; __syncthreads();
    for (int s = 128; s > 0; s >>= 1) { if (tid < s) red[tid] += red[tid + s]; __syncthreads(); }
    const float mu = red[0] / (float)T_LEN;
    __syncthreads();

    float acc2 = 0.f;
    for (int t = tid; t < T_LEN; t += 256) { float d = row[t] - mu; acc2 += d * d; }
    red[tid] = acc2; __syncthreads();
    for (int s = 128; s > 0; s >>= 1) { if (tid < s) red[tid] += red[tid + s]; __syncthreads(); }
    const float sd  = sqrtf(red[0] / (float)(T_LEN - 1));   // ddof = 1
    const float inv = 1.0f / sd;

    for (int t = tid; t < T_LEN; t += 256)
        sT[t * N_MICS + r] = (row[t] - mu) * inv;
}

// ---------------------------------------------------------------------------
// 2) gi = s.T @ w_ih.T + b_ih  via 4x chained V_WMMA_F32_16X16X4_F32
//    M=8192 (time), N=384 (gates), K=16 mics (4 K-chunks of 4).
//    Fragment layouts per CDNA5 ISA 7.12.2 (32-bit, wave32):
//      A 16x4: lanes 0-15 -> elems {0,1} = K{4c,4c+1}; lanes 16-31 -> K{4c+2,4c+3}
//      B 4x16: lanes 0-15 -> elems {0,1} = rows K{4c,4c+1}; lanes 16-31 -> K{4c+2,4c+3}
//      C/D f32: elem v -> row M = v + (lane>=16 ? 8 : 0), col N = lane&15
// ---------------------------------------------------------------------------
__global__ void gi_wmma_kernel(const float* __restrict__ sT,
                               const float* __restrict__ w_ih,
                               const float* __restrict__ b_ih,
                               float* __restrict__ gi) {
    const int wave = threadIdx.x >> 5;
    const int lane = threadIdx.x & 31;
    const int tile = blockIdx.x * (blockDim.x >> 5) + wave;
    const int tilesN = G3 / 16;                  // 24
    const int tM = tile / tilesN;
    const int tN = tile % tilesN;
    const int t0 = tM * 16;
    const int j0 = tN * 16;
    const int M  = lane & 15;                    // A-row for A, N-col for B/C/D
    const bool hi = lane >= 16;
    const int ksub = hi ? 2 : 0;                 // K sub-offset within a chunk

    const float* ap = sT  + (t0 + M) * N_MICS + ksub;   // s.T row (time t0+M)
    const float* bp = w_ih + (j0 + M) * N_MICS + ksub;  // w_ih.T col j0+M

    const float bias = b_ih[j0 + M];
    v8f C;
#pragma unroll
    for (int v = 0; v < 8; ++v) C[v] = bias;

#pragma unroll
    for (int c = 0; c < 4; ++c) {                // K = 16 in chunks of 4
        v2f A, B;
        A[0] = ap[4 * c];  A[1] = ap[4 * c + 1];
        B[0] = bp[4 * c];  B[1] = bp[4 * c + 1];
        C = __builtin_amdgcn_wmma_f32_16x16x4_f32(
                /*neg_a=*/false, A, /*neg_b=*/false, B,
                /*c_mod=*/(short)0, C, /*reuse_a=*/false, /*reuse_b=*/false);
    }

    float* g = gi + (t0 + (hi ? 8 : 0)) * G3 + j0 + M;
#pragma unroll
    for (int v = 0; v < 8; ++v) g[v * G3] = C[v];
}

// ---------------------------------------------------------------------------
// 3) Sequential GRU over 8192 steps. w_hh lives in REGISTERS: 768 threads,
//    2 per output row, 64 f32 weights each (fully unrolled). h + pair-partials
//    in a tiny LDS footprint; idle threads prefetch gi 16 rows ahead.
//    Then out = w_post @ h + b_post.
// ---------------------------------------------------------------------------
__global__ void gru_kernel(const float* __restrict__ gi,
                           const float* __restrict__ w_hh,
                           const float* __restrict__ b_hh,
                           const float* __restrict__ w_post,
                           const float* __restrict__ b_post,
                           float* __restrict__ out_vec) {
    __shared__ float h[GRU_H];
    __shared__ float part[2 * G3];          // 768 partials
    const int tid  = threadIdx.x;           // 0..767
    const int j    = tid >> 1;              // output row 0..383
    const int half = tid & 1;

    float wreg[64];
    const float* wr = w_hh + j * GRU_H + half * 64;
#pragma unroll
    for (int k = 0; k < 64; ++k) wreg[k] = wr[k];

    if (tid < GRU_H) h[tid] = 0.f;
    const float bias = half ? 0.f : b_hh[j];
    __syncthreads();

    for (int t = 0; t < T_LEN; ++t) {
        const float* hb = h + half * 64;
        float acc = bias;
#pragma unroll
        for (int k = 0; k < 64; ++k) acc = fmaf(wreg[k], hb[k], acc);
        part[tid] = acc;
        __syncthreads();
        if (tid < GRU_H) {
            const float* gt = gi + t * G3;
            const float ghr = part[2 * tid]                 + part[2 * tid + 1];
            const float ghz = part[2 * (GRU_H + tid)]       + part[2 * (GRU_H + tid) + 1];
            const float ghn = part[2 * (2 * GRU_H + tid)]   + part[2 * (2 * GRU_H + tid) + 1];
            const float r = 1.f / (1.f + expf(-(gt[tid]          + ghr)));
            const float z = 1.f / (1.f + expf(-(gt[GRU_H + tid]  + ghz)));
            const float n = tanhf(gt[2 * GRU_H + tid] + r * ghn);
            h[tid] = (1.f - z) * n + z * h[tid];
        } else if (tid >= G3 && tid < G3 + 12 && (t + 16) < T_LEN) {
            // prefetch gi row t+16 (384 floats = 12 x 128B lines) into caches
            __builtin_prefetch(gi + (t + 16) * G3 + (tid - G3) * 32, 0, 0);
        }
        __syncthreads();
    }

    if (tid < N_MICS) {
        float acc = b_post[tid];
        const float* pw = w_post + tid * GRU_H;
#pragma unroll 8
        for (int k = 0; k < GRU_H; ++k) acc = fmaf(pw[k], h[k], acc);
        out_vec[tid] = acc;
    }
}

// ---------------------------------------------------------------------------
// 4) MUSIC spectrum constant.  mf==0 => atheta==1 => spectrum[theta] is the
//    same for all theta:  1/sqrt( sum_{j<16-nsrc} (colsum of noise basis)^2 ).
//    Noise basis = first columns of Householder H = I - 2 w w^T (H u = -s e15).
//    colsum_j = 1 - 2 w_j * sum(w).
// ---------------------------------------------------------------------------
__global__ void spectrum_kernel(const float* __restrict__ out_vec,
                                const int* __restrict__ n_sources,
                                float* __restrict__ spec) {
    if (threadIdx.x != 0 || blockIdx.x != 0) return;
    float u[N_MICS];
    float nrm = 0.f;
    for (int m = 0; m < N_MICS; ++m) { u[m] = out_vec[m]; nrm += u[m] * u[m]; }
    nrm = sqrtf(nrm);
    const float inv = (nrm > 0.f) ? 1.f / nrm : 0.f;
    for (int m = 0; m < N_MICS; ++m) u[m] *= inv;
    const float s = (u[N_MICS - 1] >= 0.f) ? 1.f : -1.f;
    u[N_MICS - 1] += s;                       // w~ = u + s*e15
    float wn = 0.f;
    for (int m = 0; m < N_MICS; ++m) wn += u[m] * u[m];
    wn = sqrtf(wn);
    const float invw = (wn > 0.f) ? 1.f / wn : 0.f;
    float W = 0.f;
    for (int m = 0; m < N_MICS; ++m) { u[m] *= invw; W += u[m]; }
    const int nn = N_MICS - n_sources[0];     // 13
    float ns = 0.f;
    for (int q = 0; q < nn; ++q) { float c = 1.f - 2.f * u[q] * W; ns += c * c; }
    spec[0] = rsqrtf(ns);
}

// ---------------------------------------------------------------------------
// 5) h1[i] = gelu( spec * rowsum(w1[i,:]) + b1[i] )   (67MB streamed: the
//    HBM-roofline part, ~2.9us at 23.3 TB/s)
// ---------------------------------------------------------------------------
__global__ void h1_kernel(const float* __restrict__ w1,
                          const float* __restrict__ b1,
                          const float* __restrict__ spec,
                          float* __restrict__ h1) {
    __shared__ float red[256];
    const int i   = blockIdx.x;
    const int tid = threadIdx.x;
    const float4* row = (const float4*)(w1 + (size_t)i * N_THETAS);
    float acc = 0.f;
    for (int t = tid; t < N_THETAS / 4; t += 256) {
        float4 v = row[t];
        acc += (v.x + v.y) + (v.z + v.w);
    }
    red[tid] = acc; __syncthreads();
    for (int s = 128; s > 0; s >>= 1) { if (tid < s) red[tid] += red[tid + s]; __syncthreads(); }
    if (tid == 0) {
        float x = spec[0] * red[0] + b1[i];
        h1[i] = 0.5f * x * (1.f + erff(x * 0.70710678118654752f));   // exact gelu
    }
}

// ---------------------------------------------------------------------------
// 6) h2 = gelu(w2@h1+b2);  out = sigmoid(w3@h2+b3) * 2pi
// ---------------------------------------------------------------------------
__global__ void tail_kernel(const float* __restrict__ h1,
                            const float* __restrict__ w2,
                            const float* __restrict__ b2,
                            const float* __restrict__ w3,
                            const float* __restrict__ b3,
                            float* __restrict__ out) {
    __shared__ float s1[MLP_H], s2[MLP_H];
    const int tid = threadIdx.x;
    s1[tid] = h1[tid];
    __syncthreads();
    float acc = b2[tid];
    const float* wr = w2 + tid * MLP_H;
#pragma unroll 8
    for (int k = 0; k < MLP_H; ++k) acc = fmaf(wr[k], s1[k], acc);
    s2[tid] = 0.5f * acc * (1.f + erff(acc * 0.70710678118654752f));
    __syncthreads();
    if (tid < N_MICS) {
        float a = b3[tid];
        const float* w3r = w3 + tid * MLP_H;
#pragma unroll 8
        for (int k = 0; k < MLP_H; ++k) a = fmaf(w3r[k], s2[k], a);
        out[tid] = (1.f / (1.f + expf(-a))) * 6.283185307179586f;
    }
}

// ---------------------------------------------------------------------------
// Workspace layout (bytes, 64-aligned):
//   [0)          sT      8192*16 f32  =   524288
//   [524288)     gi      8192*384 f32 = 12582912
//   [13107200)   out_vec 16 f32
//   [13107264)   spec    1 f32 (padded)
//   [13107328)   h1      256 f32
// total ~13.11 MB
// ---------------------------------------------------------------------------
extern "C" void kernel_launch(void* const* d_in, const int* in_sizes, int n_in,
                              void* d_out, int out_size, void* d_ws, size_t ws_size,
                              hipStream_t stream) {
    (void)in_sizes; (void)n_in; (void)out_size; (void)ws_size;
    const float* samples   = (const float*)d_in[0];
    const int*   n_sources = (const int*)  d_in[1];
    // d_in[2] = mic_locations : unused (phase is exactly zero since mf == 0)
    const float* w_ih   = (const float*)d_in[3];
    const float* w_hh   = (const float*)d_in[4];
    const float* b_ih   = (const float*)d_in[5];
    const float* b_hh   = (const float*)d_in[6];
    const float* w_post = (const float*)d_in[7];
    const float* b_post = (const float*)d_in[8];
    const float* w1     = (const float*)d_in[9];
    const float* b1     = (const float*)d_in[10];
    const float* w2     = (const float*)d_in[11];
    const float* b2     = (const float*)d_in[12];
    const float* w3     = (const float*)d_in[13];
    const float* b3     = (const float*)d_in[14];

    char* ws = (char*)d_ws;
    float* sT      = (float*)(ws + 0);
    float* gi      = (float*)(ws + 524288);
    float* out_vec = (float*)(ws + 13107200);
    float* spec    = (float*)(ws + 13107264);
    float* h1      = (float*)(ws + 13107328);

    norm_kernel<<<N_MICS, 256, 0, stream>>>(samples, sT);

    // 512 * 24 = 12288 tiles, 8 waves/block -> 1536 blocks
    gi_wmma_kernel<<<1536, 256, 0, stream>>>(sT, w_ih, b_ih, gi);

    gru_kernel<<<1, 2 * G3, 0, stream>>>(gi, w_hh, b_hh, w_post, b_post, out_vec);

    spectrum_kernel<<<1, 32, 0, stream>>>(out_vec, n_sources, spec);
    h1_kernel<<<MLP_H, 256, 0, stream>>>(w1, b1, spec, h1);
    tail_kernel<<<1, MLP_H, 0, stream>>>(h1, w2, b2, w3, b3, (float*)d_out);
}